// ConvMultiHeadedAttention_23330262351899
// MI455X (gfx1250) — compile-verified
//
#include <hip/hip_runtime.h>
#include <hip/hip_bf16.h>

// ---------------------------------------------------------------------------
// ConvMultiHeadedAttention on MI455X (gfx1250, wave32, WMMA).
// GEMMs: v_wmma_f32_16x16x4_f32, 64x64 tile per wave (16 acc, 4x A/B frag reuse).
// Box softmax row staging: GLOBAL_LOAD_ASYNC_TO_LDS_B128 (ASYNCcnt path).
// ---------------------------------------------------------------------------

typedef __attribute__((ext_vector_type(2))) float v2f;
typedef __attribute__((ext_vector_type(8))) float v8f;
typedef int v4i __attribute__((vector_size(16)));   // matches builtin param type

static constexpr int B_  = 2;
static constexpr int S_  = 512;
static constexpr int D_  = 512;
static constexpr int H_  = 8;
static constexpr int DH_ = 64;
static constexpr int HS_ = 4096;                 // H*S
static constexpr long long NNZ_ = 5959864LL;     // masked slots per batch (analytic)

#if defined(__gfx1250__) && \
    __has_builtin(__builtin_amdgcn_global_load_async_to_lds_b128) && \
    __has_builtin(__builtin_amdgcn_s_wait_asynccnt)
#define USE_ASYNC_LDS 1
#else
#define USE_ASYNC_LDS 0
#endif

#if USE_ASYNC_LDS
// addrspace(1) pointer from generic VA; addrspace(3) pointer from low-32 LDS offset
#define GPTR_V4I(p) ((__attribute__((address_space(1))) v4i*)(uintptr_t)(p))
#define LPTR_V4I(p) ((__attribute__((address_space(3))) v4i*)(unsigned)(uintptr_t)(p))
#endif

// ---------------------------------------------------------------------------
// fp32 WMMA GEMM:  C = (A[M,K] @ W[N,K]^T + bias) * scale
// Wave computes a 64x64 tile: per K-step of 4, load 4 A-frags + 4 B-frags
// (8x global_load_b64) and issue 16 back-to-back v_wmma_f32_16x16x4_f32.
// Block = 128 thr (4 waves along N) -> block tile 64x256.
// mode 0: plain C[m][n] (M x 512).   mode 1: head-shuffled P[b][h*S+s][d].
// ---------------------------------------------------------------------------
__global__ void __launch_bounds__(128)
wmma_gemm_f32(const float* __restrict__ A, const float* __restrict__ W,
              const float* __restrict__ bias, float* __restrict__ C,
              int K, float scale, int mode)
{
    const int lane = threadIdx.x & 31;
    const int wave = threadIdx.x >> 5;
    const int m0 = blockIdx.x * 64;
    const int n0 = blockIdx.y * 256 + wave * 64;

    const int rc = lane & 15;            // A-frag row / B-frag col
    const int kb = (lane >> 4) * 2;      // K sub-pair within step of 4

    const float* Ap[4];
    const float* Wp[4];
#pragma unroll
    for (int t = 0; t < 4; ++t) {
        Ap[t] = A + (size_t)(m0 + 16 * t + rc) * K + kb;
        Wp[t] = W + (size_t)(n0 + 16 * t + rc) * K + kb;
    }

    v8f acc[4][4];
#pragma unroll
    for (int ar = 0; ar < 4; ++ar)
#pragma unroll
        for (int bc = 0; bc < 4; ++bc)
            acc[ar][bc] = (v8f){0.f,0.f,0.f,0.f,0.f,0.f,0.f,0.f};

    for (int kk = 0; kk < K; kk += 4) {
        v2f av[4], bv[4];
#pragma unroll
        for (int t = 0; t < 4; ++t) {
            av[t].x = Ap[t][kk]; av[t].y = Ap[t][kk + 1];
            bv[t].x = Wp[t][kk]; bv[t].y = Wp[t][kk + 1];
        }
#pragma unroll
        for (int ar = 0; ar < 4; ++ar)
#pragma unroll
            for (int bc = 0; bc < 4; ++bc)
                acc[ar][bc] = __builtin_amdgcn_wmma_f32_16x16x4_f32(
                    false, av[ar], false, bv[bc], (short)0, acc[ar][bc],
                    false, false);
    }

    const int colL = lane & 15;
    const int rAdd = 8 * (lane >> 4);
#pragma unroll
    for (int ar = 0; ar < 4; ++ar)
#pragma unroll
        for (int bc = 0; bc < 4; ++bc)
#pragma unroll
            for (int r = 0; r < 8; ++r) {
                const int row = m0 + 16 * ar + r + rAdd;
                const int col = n0 + 16 * bc + colL;
                const float v = (acc[ar][bc][r] + bias[col]) * scale;
                if (mode == 0) {
                    C[(size_t)row * D_ + col] = v;
                } else {
                    const int b = row >> 9, s = row & 511;
                    const int h = col >> 6, d = col & 63;
                    C[(((size_t)b * HS_) + h * S_ + s) * DH_ + d] = v;
                }
            }
}

// ---------------------------------------------------------------------------
// Scores + mask + exp:  E[b][i][j] = mask[b][i%S][j%S] ? 0 : exp(Qh[b,i,:]·Kh[b,j,:])
// Same 64x64-per-wave tiling; K = 64 (16 K-steps, 256 WMMAs per wave).
// ---------------------------------------------------------------------------
__global__ void __launch_bounds__(128)
scores_exp_kernel(const float* __restrict__ Qh, const float* __restrict__ Kh,
                  const unsigned char* __restrict__ mask, float* __restrict__ E)
{
    const int lane = threadIdx.x & 31;
    const int wave = threadIdx.x >> 5;
    const int b  = blockIdx.z;
    const int i0 = blockIdx.x * 64;
    const int j0 = blockIdx.y * 256 + wave * 64;

    const int rc = lane & 15;
    const int kb = (lane >> 4) * 2;

    const float* Qp[4];
    const float* Kp[4];
#pragma unroll
    for (int t = 0; t < 4; ++t) {
        Qp[t] = Qh + ((size_t)b * HS_ + i0 + 16 * t + rc) * DH_ + kb;
        Kp[t] = Kh + ((size_t)b * HS_ + j0 + 16 * t + rc) * DH_ + kb;
    }

    v8f acc[4][4];
#pragma unroll
    for (int ar = 0; ar < 4; ++ar)
#pragma unroll
        for (int bc = 0; bc < 4; ++bc)
            acc[ar][bc] = (v8f){0.f,0.f,0.f,0.f,0.f,0.f,0.f,0.f};

    for (int kk = 0; kk < DH_; kk += 4) {
        v2f av[4], bv[4];
#pragma unroll
        for (int t = 0; t < 4; ++t) {
            av[t].x = Qp[t][kk]; av[t].y = Qp[t][kk + 1];
            bv[t].x = Kp[t][kk]; bv[t].y = Kp[t][kk + 1];
        }
#pragma unroll
        for (int ar = 0; ar < 4; ++ar)
#pragma unroll
            for (int bc = 0; bc < 4; ++bc)
                acc[ar][bc] = __builtin_amdgcn_wmma_f32_16x16x4_f32(
                    false, av[ar], false, bv[bc], (short)0, acc[ar][bc],
                    false, false);
    }

    const int colL = lane & 15;
    const int rAdd = 8 * (lane >> 4);
#pragma unroll
    for (int ar = 0; ar < 4; ++ar)
#pragma unroll
        for (int bc = 0; bc < 4; ++bc)
#pragma unroll
            for (int r = 0; r < 8; ++r) {
                const int i = i0 + 16 * ar + r + rAdd;
                const int j = j0 + 16 * bc + colL;
                const int q = i & 511, k = j & 511;
                const unsigned char m = mask[((size_t)b * S_ + q) * S_ + k];
                const float e = m ? 0.0f : __expf(acc[ar][bc][r]);
                E[((size_t)b * HS_ + i) * HS_ + j] = e;
            }
}

// ---------------------------------------------------------------------------
// In-place row-local box softmax.  One block per (b, i) row (8192 rows).
// Row staging uses GLOBAL_LOAD_ASYNC_TO_LDS_B128 (ASYNCcnt) when available.
// Also emits top_attn = attn[:,0,:,0,:] for rows with hi==0.
// ---------------------------------------------------------------------------
__global__ void __launch_bounds__(256)
box_softmax_kernel(float* __restrict__ E, float* __restrict__ top)
{
    __shared__ float er[HS_];
    __shared__ float s1[HS_];
    const int tid = threadIdx.x;
    const int row = blockIdx.x;          // b*4096 + i
    const int b  = row >> 12;
    const int ii = row & 4095;
    const int hi = ii >> 9;
    const int q  = ii & 511;
    float* Erow = E + ((size_t)b * HS_ + ii) * HS_;

#if USE_ASYNC_LDS
    // 16KB row -> LDS via async DMA path: 256 thr x 4 passes x b128.
    for (int t = tid * 4; t < HS_; t += 256 * 4) {
        __builtin_amdgcn_global_load_async_to_lds_b128(
            GPTR_V4I(Erow + t), LPTR_V4I(er + t), 0, 0);
    }
    __builtin_amdgcn_s_wait_asynccnt(0);
#else
    for (int t = tid; t < HS_; t += 256) er[t] = Erow[t];
#endif
    __syncthreads();

    for (int t = tid; t < HS_; t += 256) {
        const int hj = t >> 9, k = t & 511;
        const int lo = (k - 4 < 0) ? 0 : k - 4;
        const int hi2 = (k + 4 > 511) ? 511 : k + 4;
        float s = 0.f;
        for (int kk = lo; kk <= hi2; ++kk) s += er[(hj << 9) + kk];
        s1[t] = s;
    }
    __syncthreads();

    for (int t = tid; t < HS_; t += 256) {
        const int hj = t >> 9;
        float d = s1[t];
        if (hj > 0) d += s1[t - 512];
        if (hj < 7) d += s1[t + 512];
        const float a = er[t] / d;
        Erow[t] = a;
        if (hi == 0 && t < 512)
            top[((size_t)b * S_ + q) * S_ + t] = a;
    }
}

// ---------------------------------------------------------------------------
// ctx = attn_masked @ V with torch masked_scatter semantics, fused with the
// (B,dh,h*S)->(B,S,D) reshape.  mw = (|hi-hj|<=1) | (|q-k|<=4) is analytic, so
// the gather source E.flat[b*NNZ + rank] is a CONTIGUOUS run per row.
// One wave per row; lanes hold d-pairs.
// ---------------------------------------------------------------------------
__global__ void __launch_bounds__(256)
ctx_scatter_kernel(const float* __restrict__ E, const float* __restrict__ Vh,
                   float* __restrict__ Xr)
{
    const int lane = threadIdx.x & 31;
    const int wid  = (blockIdx.x * blockDim.x + threadIdx.x) >> 5;  // 0..8191
    const int b  = wid >> 12;
    const int ii = wid & 4095;
    const int hi = ii >> 9;
    const int q  = ii & 511;

    // rank of first masked slot in this row (batch-local)
    long long start = 0;
    for (int h2 = 0; h2 < hi; ++h2) {
        const int nh = (h2 == 0 || h2 == 7) ? 2 : 3;
        start += (long long)nh * 262144LL + (long long)(8 - nh) * 4588LL;
    }
    int Wpre = 0;
    for (int qp = 0; qp < q; ++qp) {
        const int lo = (qp - 4 < 0) ? 0 : qp - 4;
        const int hi2 = (qp + 4 > 511) ? 511 : qp + 4;
        Wpre += hi2 - lo + 1;
    }
    const int nHi = (hi == 0 || hi == 7) ? 2 : 3;
    start += (long long)q * 512LL * nHi + (long long)(8 - nHi) * (long long)Wpre;

    // global flat gather base (masked_scatter consumes attn flat ACROSS batches)
    const float* src = E + (long long)b * NNZ_ + start;

    float ax = 0.f, ay = 0.f;
    int t = 0;
    for (int hj = 0; hj < 8; ++hj) {
        const int dh2 = (hi > hj) ? hi - hj : hj - hi;
        int k0, k1;
        if (dh2 <= 1) { k0 = 0; k1 = 511; }
        else { k0 = (q - 4 < 0) ? 0 : q - 4; k1 = (q + 4 > 511) ? 511 : q + 4; }
        __builtin_prefetch(src + t, 0, 1);   // contiguous stream -> global_prefetch_b8
        const float* vrow = Vh + ((size_t)b * HS_ + (hj << 9) + k0) * DH_ + 2 * lane;
        for (int k = k0; k <= k1; ++k) {
            const float a = src[t++];        // uniform across lanes (broadcast)
            ax += a * vrow[0];
            ay += a * vrow[1];
            vrow += DH_;
        }
    }
    // write in the reshaped layout: out-row = d*8 + hi, out-col = q
    const int d0 = 2 * lane;
    Xr[((size_t)b * S_ + (d0 * 8 + hi)) * D_ + q]       = ax;
    Xr[((size_t)b * S_ + ((d0 + 1) * 8 + hi)) * D_ + q] = ay;
}

// ---------------------------------------------------------------------------
// Launch: proj(Q,K,V) -> scores+exp -> box softmax (in place, +top_attn)
//         -> banded scatter-SpMM (reshaped ctx) -> output GEMM.
// ---------------------------------------------------------------------------
extern "C" void kernel_launch(void* const* d_in, const int* in_sizes, int n_in,
                              void* d_out, int out_size, void* d_ws, size_t ws_size,
                              hipStream_t stream)
{
    const float* key_in   = (const float*)d_in[0];
    const float* value_in = (const float*)d_in[1];
    const float* query_in = (const float*)d_in[2];
    const unsigned char* mask = (const unsigned char*)d_in[3];  // jnp.bool_ = 1 byte
    const float* Wq = (const float*)d_in[4];
    const float* bq = (const float*)d_in[5];
    const float* Wk = (const float*)d_in[6];
    const float* bk = (const float*)d_in[7];
    const float* Wv = (const float*)d_in[8];
    const float* bv = (const float*)d_in[9];
    const float* Wo = (const float*)d_in[10];
    const float* bo = (const float*)d_in[11];

    // workspace partition (floats): Qh | Kh | Vh | Xr | E  (~143 MB total)
    float* wsf = (float*)d_ws;
    float* Qh = wsf;                     // B*HS*DH = 524288
    float* Kh = wsf + 524288;
    float* Vh = wsf + 1048576;
    float* Xr = wsf + 1572864;           // B*S*D   = 524288
    float* E  = wsf + 2097152;           // B*HS*HS = 33554432

    float* out = (float*)d_out;          // (B,S,D)
    float* top = out + 524288;           // (B,S,S)

    const dim3 gp(16, 2, 1);             // M/64 x N/256 tiles, 128 thr (4 waves)
    wmma_gemm_f32<<<gp, 128, 0, stream>>>(query_in, Wq, bq, Qh, D_, 0.125f, 1);
    wmma_gemm_f32<<<gp, 128, 0, stream>>>(key_in,   Wk, bk, Kh, D_, 1.0f,   1);
    wmma_gemm_f32<<<gp, 128, 0, stream>>>(value_in, Wv, bv, Vh, D_, 1.0f,   1);

    scores_exp_kernel<<<dim3(64, 16, 2), 128, 0, stream>>>(Qh, Kh, mask, E);
    box_softmax_kernel<<<8192, 256, 0, stream>>>(E, top);
    ctx_scatter_kernel<<<1024, 256, 0, stream>>>(E, Vh, Xr);

    wmma_gemm_f32<<<gp, 128, 0, stream>>>(Xr, Wo, bo, out, D_, 1.0f, 0);
}